// SampleGroupMoudle_part_51144470561303
// MI455X (gfx1250) — compile-verified
//
#include <hip/hip_runtime.h>

// Farthest Point Sampling, MI455X (gfx1250, wave32).
// - One 1024-thread workgroup (32 wave32 waves) per batch, 64 pts/thread.
// - Running min-distance is register-resident (64 f32/thread) -> zero memory
//   traffic for the recurrence state.
// - 26624 points cached in LDS as packed 12B float3 (319.5KB of the 320KB WGP
//   LDS, probe-confirmed to compile): stride-12 b32 triplets are bank-conflict
//   free (3*lane mod 64 is a bijection over wave32).
// - Remaining points streamed from a dense float3 copy in d_ws (one-time pack
//   kernel, global_load_b96/global_store_b96): 100% cacheline utilization,
//   dataset (24MB) stays resident in the 192MB L2.
// - Argmax: per-thread scan (ascending index + strict '>' keeps first
//   occurrence) -> wave32 XOR butterfly -> one ds_max_u64 per wave leader on a
//   packed (value_bits<<32)|~index key: non-negative f32 bit patterns are
//   monotone as u32, and ~index makes ties resolve to the LOWEST index,
//   matching jnp.argmax exactly.
// - Output dtype: harness materializes non-bf16 outputs as float32, so indices
//   (< 65536, exactly representable) are written as floats.

#define NBATCH  32
#define NPTS    65536
#define NCH     6
#define NPOINT  1024
#define TPB     1024
#define PPT     (NPTS / TPB)      // 64 points per thread
#define LDSJ    26                // j-slots cached in LDS
#define LDS_N   (LDSJ * TPB)      // 26624 points -> 319,488 B of LDS

struct __align__(4) P3 { float x, y, z; };

template <int S>   // source stride in floats: 3 = packed ws, 6 = raw input
__global__ __launch_bounds__(TPB) void fps_kernel(const float* __restrict__ src,
                                                  float* __restrict__ out)
{
    __shared__ P3 spts[LDS_N];                 // 319,488 B
    __shared__ unsigned long long slot[2];     // double-buffered argmax key

    const int b   = blockIdx.x;
    const int tid = threadIdx.x;
    const float* __restrict__ sb = src + (size_t)b * NPTS * S;

    if (tid == 0) { slot[0] = 0ull; slot[1] = 0ull; }
    for (int p = tid; p < LDS_N; p += TPB) {
        const float* q = sb + (size_t)p * S;
        P3 v; v.x = q[0]; v.y = q[1]; v.z = q[2];
        spts[p] = v;
    }
    __syncthreads();

    float dist[PPT];
#pragma unroll
    for (int j = 0; j < PPT; ++j) dist[j] = 1e10f;

    int far = 0;

    for (int it = 0; it < NPOINT; ++it) {
        if (tid == 0) out[b * NPOINT + it] = (float)far;  // emit carry (scan)

        // Centroid: uniform address -> broadcast through cache.
        const float* c = sb + (size_t)far * S;
        const float cx = c[0], cy = c[1], cz = c[2];

        float bv = -1.0f;
        int   bi = 0;

        // LDS-resident points (conflict-free b32 triplets).
#pragma unroll
        for (int j = 0; j < LDSJ; ++j) {
            const int p = j * TPB + tid;
            const P3 v = spts[p];
            const float dx = v.x - cx, dy = v.y - cy, dz = v.z - cz;
            const float d  = dx * dx + dy * dy + dz * dz;
            const float nd = fminf(dist[j], d);
            dist[j] = nd;
            if (nd > bv) { bv = nd; bi = p; }   // strict > keeps lowest index
        }
        // L2-streamed points: dense float3 -> b96 loads, full line utilization.
#pragma unroll
        for (int j = LDSJ; j < PPT; ++j) {
            const int p = j * TPB + tid;
            const float* q = sb + (size_t)p * S;
            const float dx = q[0] - cx, dy = q[1] - cy, dz = q[2] - cz;
            const float d  = dx * dx + dy * dy + dz * dz;
            const float nd = fminf(dist[j], d);
            dist[j] = nd;
            if (nd > bv) { bv = nd; bi = p; }
        }

        // wave32 XOR-butterfly argmax (max value, tie -> lowest index).
#pragma unroll
        for (int m = 16; m > 0; m >>= 1) {
            const float ov = __shfl_xor(bv, m, 32);
            const int   oi = __shfl_xor(bi, m, 32);
            if (ov > bv || (ov == bv && oi < bi)) { bv = ov; bi = oi; }
        }

        // Cross-wave reduce: 32 leaders ds_max_u64 a packed key.
        const int A = it & 1;
        if ((tid & 31) == 0) {
            const unsigned long long key =
                ((unsigned long long)__float_as_uint(bv) << 32) |
                (unsigned int)(~bi);
            atomicMax(&slot[A], key);
        }
        __syncthreads();

        far = (int)(~(unsigned int)slot[A]);    // broadcast winner
        if (tid == 0) slot[A ^ 1] = 0ull;       // dead slot, safe to reset here
        __syncthreads();
    }
}

__global__ void pack_kernel(const float* __restrict__ x, P3* __restrict__ w) {
    const size_t i = (size_t)blockIdx.x * blockDim.x + threadIdx.x;
    if (i < (size_t)NBATCH * NPTS) {
        const float* q = x + i * NCH;
        P3 p; p.x = q[0]; p.y = q[1]; p.z = q[2];
        w[i] = p;
    }
}

extern "C" void kernel_launch(void* const* d_in, const int* in_sizes, int n_in,
                              void* d_out, int out_size, void* d_ws, size_t ws_size,
                              hipStream_t stream) {
    (void)in_sizes; (void)n_in; (void)out_size;
    const float* x = (const float*)d_in[0];
    float* out = (float*)d_out;   // non-bf16 outputs are float32 per harness

    const size_t need = (size_t)NBATCH * NPTS * sizeof(P3);   // 24 MB
    if (ws_size >= need) {
        const int total = NBATCH * NPTS;
        pack_kernel<<<(total + 255) / 256, 256, 0, stream>>>(x, (P3*)d_ws);
        fps_kernel<3><<<NBATCH, TPB, 0, stream>>>((const float*)d_ws, out);
    } else {
        fps_kernel<6><<<NBATCH, TPB, 0, stream>>>(x, out);
    }
}